// Dy_UpSample_30880814858243
// MI455X (gfx1250) — compile-verified
//
#include <hip/hip_runtime.h>
#include <hip/hip_bf16.h>

typedef __attribute__((ext_vector_type(2))) float v2f;
typedef __attribute__((ext_vector_type(8))) float v8f;

#define Cin   256
#define HW    65536      // 256*256
#define HWo   262144     // 512*512

// ---------------------------------------------------------------------------
// Kernel A: fused 1x1-conv pair via f32 WMMA.
//   y[c, p] = sum_k W[c,k] * x[k, p]   for c in [0,64)  (0..31 = off, 32..63 = scope)
//   o[c, p] = sigmoid(y[c+32]+b_scope[c]) * 0.5 * (y[c]+b_off[c]),  c in [0,32)
// Block = 256 threads (8 waves); block owns 32 pixels; wave = one 16x16 tile.
// ---------------------------------------------------------------------------
__global__ __launch_bounds__(256) void dysample_offsets_kernel(
    const float* __restrict__ x,       // [256, 65536]
    const float* __restrict__ w_off,   // [32, 256]
    const float* __restrict__ b_off,   // [32]
    const float* __restrict__ w_scope, // [32, 256]
    const float* __restrict__ b_scope, // [32]
    float* __restrict__ o_out)         // [32, 65536]
{
    __shared__ float sX[Cin * 32];   // [k][p], 32 KB
    __shared__ float sY[64 * 32];    // [c][p],  8 KB

    const int tid   = threadIdx.x;
    const int pbase = blockIdx.x * 32;

    // Stage x tile: 8192 floats, coalesced b128 loads (each k-row = 128 B).
    for (int i = tid; i < 2048; i += 256) {
        const int k = i >> 3;
        const int q = (i & 7) << 2;
        const float4 v = *(const float4*)(x + (size_t)k * HW + pbase + q);
        *(float4*)(&sX[k * 32 + q]) = v;
    }
    __syncthreads();

    const int lane = tid & 31;
    const int wv   = tid >> 5;
    const int mt   = wv & 3;        // M tile: 0,1 -> off rows; 2,3 -> scope rows
    const int nt   = wv >> 2;       // N tile: 0,1
    const int row  = lane & 15;     // M (for A) / N (for B,C) index
    const int hi   = lane >> 4;     // lane half
    const int pxb  = nt * 16;

    const float* Wg   = (mt < 2) ? w_off : w_scope;
    const float* wrow = Wg + ((mt & 1) * 16 + row) * Cin;   // this lane's weight row

    v8f acc = {};
    #pragma unroll 4
    for (int k0 = 0; k0 < Cin; k0 += 4) {
        const int kk = k0 + 2 * hi;
        // A (16x4 f32): lanes 0-15 hold K=k0,k0+1 ; lanes 16-31 hold K=k0+2,k0+3
        v2f a = *(const v2f*)(wrow + kk);                   // global b64, row-contiguous
        // B (4x16 f32): same K split; N = lane%16
        v2f b;
        b.x = sX[kk * 32 + pxb + row];
        b.y = sX[(kk + 1) * 32 + pxb + row];
        acc = __builtin_amdgcn_wmma_f32_16x16x4_f32(
            /*neg_a=*/false, a, /*neg_b=*/false, b,
            /*c_mod=*/(short)0, acc, /*reuse_a=*/false, /*reuse_b=*/false);
    }

    // C layout: VGPR j, lane L -> M = j + 8*(L>=16), N = L%16
    #pragma unroll
    for (int j = 0; j < 8; ++j) {
        const int c = mt * 16 + j + 8 * hi;   // 0..63 across the 4 M-tiles
        sY[c * 32 + pxb + row] = acc[j];
    }
    __syncthreads();

    // Fuse: bias + sigmoid(scope) * 0.5 * off  ->  o[c, p]
    for (int i = tid; i < 1024; i += 256) {
        const int c = i >> 5;        // 0..31
        const int p = i & 31;
        const float yo = sY[c * 32 + p] + b_off[c];
        const float ys = sY[(c + 32) * 32 + p] + b_scope[c];
        const float s  = 1.0f / (1.0f + __expf(-ys));
        o_out[(size_t)c * HW + pbase + p] = s * 0.5f * yo;
    }
}

// ---------------------------------------------------------------------------
// Kernel B: offset combine + grid_sample (border, bilinear), 64 channels/thread.
// Thread -> (g, yy, xx); lanes span consecutive xx so all 64 stores are
// 128B-coalesced per wave.  x gathers are strided (reference swaps row<->x
// coord) but x is fully L2-resident (64 MB < 192 MB).
// ---------------------------------------------------------------------------
__global__ __launch_bounds__(256) void dysample_sample_kernel(
    const float* __restrict__ x,    // [256, 256, 256]
    const float* __restrict__ off,  // [256, 256, 2]  (H, W, t)
    const float* __restrict__ o,    // [32, 65536]    (from kernel A)
    float* __restrict__ out)        // [256, 512, 512]
{
    const int gid = blockIdx.x * 256 + threadIdx.x;
    const int xx  = gid & 511;
    const int yy  = (gid >> 9) & 511;
    const int g   = gid >> 18;                  // 0..3

    // Pixel-shuffle read of o: offset[g*2+t, yy, xx] = o[(g*2+t)*4 + ry*2+rx, y, x]
    const int yc = yy >> 1, ry = yy & 1;
    const int xc = xx >> 1, rx = xx & 1;
    const int sub = ry * 2 + rx;
    const int cp  = yc * 256 + xc;
    const float og0 = o[(size_t)((g * 2 + 0) * 4 + sub) * HW + cp];
    const float og1 = o[(size_t)((g * 2 + 1) * 4 + sub) * HW + cp];

    // Bilinear 2x upsample of `off` (align_corners=False): s = max(i/2 - 0.25, 0)
    const float sy = fmaxf((float)yy * 0.5f - 0.25f, 0.0f);
    const float sx = fmaxf((float)xx * 0.5f - 0.25f, 0.0f);
    const int y0 = (int)sy;  const int y1 = min(y0 + 1, 255);  const float fy = sy - (float)y0;
    const int x0 = (int)sx;  const int x1 = min(x0 + 1, 255);  const float fx = sx - (float)x0;
    const float* p00 = off + ((size_t)y0 * 256 + x0) * 2;
    const float* p01 = off + ((size_t)y0 * 256 + x1) * 2;
    const float* p10 = off + ((size_t)y1 * 256 + x0) * 2;
    const float* p11 = off + ((size_t)y1 * 256 + x1) * 2;
    const float up0 = (1.0f - fy) * ((1.0f - fx) * p00[0] + fx * p01[0])
                    +         fy  * ((1.0f - fx) * p10[0] + fx * p11[0]);
    const float up1 = (1.0f - fy) * ((1.0f - fx) * p00[1] + fx * p01[1])
                    +         fy  * ((1.0f - fx) * p10[1] + fx * p11[1]);

    const float oc0 = 0.5f * (up0 + og0);
    const float oc1 = 0.5f * (up1 + og1);

    // grid = 2*(pos+offc)/norm - 1 with pos0=row, used as sample-x (reference quirk):
    //   ix = (row + 0.5 + oc0)*W/w - 0.5 ; iy = (col + 0.5 + oc1)*H/h - 0.5 ; border clamp
    float ix = fminf(fmaxf(0.5f * ((float)yy + 0.5f + oc0) - 0.5f, 0.0f), 255.0f);
    float iy = fminf(fmaxf(0.5f * ((float)xx + 0.5f + oc1) - 0.5f, 0.0f), 255.0f);

    const float ix0f = floorf(ix); const float tx = ix - ix0f;
    const float iy0f = floorf(iy); const float ty = iy - iy0f;
    const int ix0 = (int)ix0f; const int ix1 = min(ix0 + 1, 255);
    const int iy0 = (int)iy0f; const int iy1 = min(iy0 + 1, 255);

    const float w00 = (1.0f - tx) * (1.0f - ty);
    const float w01 =         tx  * (1.0f - ty);
    const float w10 = (1.0f - tx) *         ty;
    const float w11 =         tx  *         ty;

    const int b00 = iy0 * 256 + ix0;
    const int b01 = iy0 * 256 + ix1;
    const int b10 = iy1 * 256 + ix0;
    const int b11 = iy1 * 256 + ix1;
    const int ob  = yy * 512 + xx;

    // Group g holds source channels c = 4*c2 + g; output channel is the same c.
    #pragma unroll 4
    for (int c2 = 0; c2 < 64; ++c2) {
        const int c = 4 * c2 + g;
        const float* pc = x + (size_t)c * HW;
        const float v = w00 * pc[b00] + w01 * pc[b01] + w10 * pc[b10] + w11 * pc[b11];
        out[(size_t)c * HWo + ob] = v;
    }
}

// ---------------------------------------------------------------------------
// Inputs (setup_inputs order): x, off, w_off, b_off, w_scope, b_scope
// ---------------------------------------------------------------------------
extern "C" void kernel_launch(void* const* d_in, const int* in_sizes, int n_in,
                              void* d_out, int out_size, void* d_ws, size_t ws_size,
                              hipStream_t stream) {
    const float* x       = (const float*)d_in[0];
    const float* off     = (const float*)d_in[1];
    const float* w_off   = (const float*)d_in[2];
    const float* b_off   = (const float*)d_in[3];
    const float* w_scope = (const float*)d_in[4];
    const float* b_scope = (const float*)d_in[5];
    float* out  = (float*)d_out;
    float* o_ws = (float*)d_ws;    // 32 * 65536 floats = 8 MB scratch

    // Kernel A: 65536 pixels / 32 per block
    dysample_offsets_kernel<<<2048, 256, 0, stream>>>(x, w_off, b_off, w_scope, b_scope, o_ws);
    // Kernel B: 4 groups * 512 * 512 threads
    dysample_sample_kernel<<<4096, 256, 0, stream>>>(x, off, o_ws, out);
}